// GCN_50302656971586
// MI455X (gfx1250) — compile-verified
//
#include <hip/hip_runtime.h>
#include <hip/hip_bf16.h>

// ---------------------------------------------------------------------------
// 2-layer GCN on MI455X (gfx1250, wave32).
//   H0 = x @ W1                       (WMMA f32 16x16x4)
//   H1 = relu(spmm(adj, H0) + b1)     (edge-parallel f32 atomics, L2-resident)
//   G2 = H1 @ W2                      (WMMA f32 16x16x4, N padded 40->48)
//   O2 = spmm(adj, G2)
//   out = log_softmax(O2[idx] + b2)   (fused bias + softmax on 2048 rows only)
// ---------------------------------------------------------------------------

typedef float v2f __attribute__((ext_vector_type(2)));
typedef float v8f __attribute__((ext_vector_type(8)));

constexpr int N_NODES = 50000;
constexpr int N_EDGES = 800000;
constexpr int NFEAT   = 256;
constexpr int NHID    = 128;
constexpr int NCLASS  = 40;
constexpr int N_IDX   = 2048;
constexpr int MTILES1 = N_NODES / 16;   // 3125, exact

// ---------------------------------------------------------------------------
// GEMM1: H0[50000x128] = X[50000x256] @ W1[256x128]
// Block = 256 threads = 8 waves; one 16-row M tile per block; wave w owns
// N-tile w (NHID = 8*16). A tile staged in LDS; W1 (128 KB) stays L2-hot.
// ---------------------------------------------------------------------------
__global__ __launch_bounds__(256)
void gemm1_wmma(const float* __restrict__ X, const float* __restrict__ W,
                float* __restrict__ H0) {
    __shared__ float xs[16 * NFEAT];                 // 16 KB
    const int mbase = blockIdx.x * 16;
    const int tid   = threadIdx.x;

    // Cooperative load of the 16x256 A tile (coalesced, 16 floats/thread).
    for (int i = tid; i < 16 * NFEAT; i += 256) {
        xs[i] = X[(mbase + (i >> 8)) * NFEAT + (i & (NFEAT - 1))];
    }
    __syncthreads();

    const int wave = tid >> 5;            // 0..7 -> N tile
    const int lane = tid & 31;
    const int half = lane >> 4;           // 0: K=0,1   1: K=2,3 (ISA A/B layout)
    const int l16  = lane & 15;
    const int n    = wave * 16 + l16;     // output column

    v8f acc = {};
    for (int k = 0; k < NFEAT; k += 4) {
        const int kk = k + half * 2;
        v2f a, b;
        a.x = xs[l16 * NFEAT + kk];
        a.y = xs[l16 * NFEAT + kk + 1];
        b.x = W[kk * NHID + n];
        b.y = W[(kk + 1) * NHID + n];
        acc = __builtin_amdgcn_wmma_f32_16x16x4_f32(
            /*neg_a=*/false, a, /*neg_b=*/false, b,
            /*c_mod=*/(short)0, acc, /*reuse_a=*/false, /*reuse_b=*/false);
    }

    // C layout: VGPR r -> M = r (lanes 0-15) / M = 8+r (lanes 16-31), N = l16.
    const int mrow = mbase + half * 8;
#pragma unroll
    for (int r = 0; r < 8; ++r) {
        H0[(mrow + r) * NHID + n] = acc[r];
    }
}

// ---------------------------------------------------------------------------
// GEMM2: G2[50000x40] = H1[50000x128] @ W2[128x40]
// grid = (ceil(3125/4), 3 N-tiles), block = 128 (4 waves, one M tile each).
// Column guard pads N to 48; guards are lane-level cndmask, EXEC full at WMMA.
// ---------------------------------------------------------------------------
__global__ __launch_bounds__(128)
void gemm2_wmma(const float* __restrict__ H1, const float* __restrict__ W2,
                float* __restrict__ G2) {
    const int wave  = threadIdx.x >> 5;
    const int mtile = blockIdx.x * 4 + wave;
    if (mtile >= MTILES1) return;                    // wave-uniform branch
    const int lane  = threadIdx.x & 31;
    const int half  = lane >> 4;
    const int l16   = lane & 15;
    const int col   = blockIdx.y * 16 + l16;
    const int mbase = mtile * 16;

    v8f acc = {};
    for (int k = 0; k < NHID; k += 4) {
        const int kk = k + half * 2;
        v2f a, b;
        a.x = H1[(mbase + l16) * NHID + kk];
        a.y = H1[(mbase + l16) * NHID + kk + 1];
        b.x = (col < NCLASS) ? W2[kk * NCLASS + col] : 0.0f;
        b.y = (col < NCLASS) ? W2[(kk + 1) * NCLASS + col] : 0.0f;
        acc = __builtin_amdgcn_wmma_f32_16x16x4_f32(
            false, a, false, b, (short)0, acc, false, false);
    }

    if (col < NCLASS) {
        const int mrow = mbase + half * 8;
#pragma unroll
        for (int r = 0; r < 8; ++r) {
            G2[(mrow + r) * NCLASS + col] = acc[r];
        }
    }
}

// ---------------------------------------------------------------------------
// SpMM: out[dst[e], f] += ew[e] * H[src[e], f].  blockDim must be (edges/blk)*F.
// H fits in the 192MB L2 (25.6MB / 8MB), so gathers are L2-resident.
// ---------------------------------------------------------------------------
__global__ void spmm_atomic(const float* __restrict__ H, const float* __restrict__ ew,
                            const int* __restrict__ src, const int* __restrict__ dst,
                            float* __restrict__ out, int nEdges, int F) {
    const int epb = blockDim.x / F;                  // edges per block (exact)
    const int le  = threadIdx.x / F;
    const int f   = threadIdx.x - le * F;
    const int e   = blockIdx.x * epb + le;
    if (e >= nEdges) return;
    const int s = src[e];
    const int d = dst[e];
    const float v = ew[e] * H[s * F + f];
    atomicAdd(&out[d * F + f], v);                   // global_atomic_add_f32
}

// ---------------------------------------------------------------------------
// H1 = relu(H1 + b1) elementwise (F = NHID = 128, power of two).
// ---------------------------------------------------------------------------
__global__ void bias_relu(float* __restrict__ H, const float* __restrict__ b, int n) {
    const int i = blockIdx.x * blockDim.x + threadIdx.x;
    if (i < n) {
        const float v = H[i] + b[i & (NHID - 1)];
        H[i] = v > 0.0f ? v : 0.0f;
    }
}

__global__ void fill_zero(float* __restrict__ p, int n) {
    const int i = blockIdx.x * blockDim.x + threadIdx.x;
    if (i < n) p[i] = 0.0f;
}

// ---------------------------------------------------------------------------
// out[j, :] = log_softmax(O2[idx[j], :] + b2).  One wave32 per row; lanes hold
// classes {lane, lane+32}; shuffle-reduce max then sum(exp).
// ---------------------------------------------------------------------------
__global__ __launch_bounds__(256)
void logsoftmax_gather(const float* __restrict__ O2, const float* __restrict__ b2,
                       const int* __restrict__ idx, float* __restrict__ out, int nIdx) {
    const int wave = threadIdx.x >> 5;
    const int lane = threadIdx.x & 31;
    const int j    = blockIdx.x * 8 + wave;
    if (j >= nIdx) return;
    const int row = idx[j];

    const int c1 = lane + 32;
    float v0 = (lane < NCLASS) ? O2[row * NCLASS + lane] + b2[lane] : -INFINITY;
    float v1 = (c1 < NCLASS)   ? O2[row * NCLASS + c1]   + b2[c1]   : -INFINITY;

    float m = fmaxf(v0, v1);
#pragma unroll
    for (int off = 16; off > 0; off >>= 1) m = fmaxf(m, __shfl_xor(m, off, 32));

    float s = 0.0f;
    if (lane < NCLASS) s += expf(v0 - m);
    if (c1 < NCLASS)   s += expf(v1 - m);
#pragma unroll
    for (int off = 16; off > 0; off >>= 1) s += __shfl_xor(s, off, 32);

    const float lse = m + logf(s);
    if (lane < NCLASS) out[j * NCLASS + lane] = v0 - lse;
    if (c1 < NCLASS)   out[j * NCLASS + c1]   = v1 - lse;
}

// ---------------------------------------------------------------------------
extern "C" void kernel_launch(void* const* d_in, const int* in_sizes, int n_in,
                              void* d_out, int out_size, void* d_ws, size_t ws_size,
                              hipStream_t stream) {
    const float* x   = (const float*)d_in[0];
    const float* W1  = (const float*)d_in[1];
    const float* b1  = (const float*)d_in[2];
    const float* W2  = (const float*)d_in[3];
    const float* b2  = (const float*)d_in[4];
    const float* ew  = (const float*)d_in[5];
    const int*   src = (const int*)d_in[6];
    const int*   dst = (const int*)d_in[7];
    const int*   idx = (const int*)d_in[8];
    float* out = (float*)d_out;

    // Workspace layout (59.2 MB): H0 | H1 | O2 ; G2 aliases H0 (dead after spmm1).
    float* H0 = (float*)d_ws;
    float* H1 = H0 + (size_t)N_NODES * NHID;
    float* O2 = H1 + (size_t)N_NODES * NHID;
    float* G2 = H0;

    const int nH1 = N_NODES * NHID;     // 6.4M
    const int nO2 = N_NODES * NCLASS;   // 2.0M

    // Zero the scatter-add targets (deterministic per-call init).
    fill_zero<<<(nH1 + 255) / 256, 256, 0, stream>>>(H1, nH1);
    fill_zero<<<(nO2 + 255) / 256, 256, 0, stream>>>(O2, nO2);

    // H0 = x @ W1
    gemm1_wmma<<<MTILES1, 256, 0, stream>>>(x, W1, H0);

    // H1 = spmm(H0);  2 edges per 256-thread block (F=128)
    spmm_atomic<<<N_EDGES / 2, 256, 0, stream>>>(H0, ew, src, dst, H1, N_EDGES, NHID);

    // H1 = relu(H1 + b1)
    bias_relu<<<(nH1 + 255) / 256, 256, 0, stream>>>(H1, b1, nH1);

    // G2 = H1 @ W2
    dim3 g2((MTILES1 + 3) / 4, 3);
    gemm2_wmma<<<g2, 128, 0, stream>>>(H1, W2, G2);

    // O2 = spmm(G2);  4 edges per 160-thread block (F=40)
    spmm_atomic<<<N_EDGES / 4, 160, 0, stream>>>(G2, ew, src, dst, O2, N_EDGES, NCLASS);

    // out = log_softmax(O2[idx] + b2)
    logsoftmax_gather<<<(N_IDX + 7) / 8, 256, 0, stream>>>(O2, b2, idx, out, N_IDX);
}